// GemsNetT_37563783971147
// MI455X (gfx1250) — compile-verified
//
#include <hip/hip_runtime.h>
#include <math.h>

#define INV_SQRT2F 0.70710678118654752440f

typedef __attribute__((ext_vector_type(2))) float v2f;
typedef __attribute__((ext_vector_type(8))) float v8f;

__device__ __forceinline__ float siluf(float x) { return x / (1.0f + __expf(-x)); }

// ------------------------------------------------------------------
// WMMA fp32 GEMM, Nn multiple of 128. grid=(ceil(M/16), Nn/128), 256 thr.
// Each wave computes one 16x16 tile via v_wmma_f32_16x16x4_f32.
// ------------------------------------------------------------------
template<bool ACT>
__global__ void k_gemm_nb(const float* __restrict__ A, const float* __restrict__ W,
                          float* __restrict__ C, int M, int K, int Nn, int ldc) {
  __shared__ float sA[16 * 513]; // K up to 512, padded stride
  const int m0   = blockIdx.x * 16;
  const int wave = threadIdx.x >> 5;
  const int lane = threadIdx.x & 31;
  const int n0   = blockIdx.y * 128 + wave * 16;
  const int ldsK = K + 1;
  for (int r = 0; r < 16; ++r) {
    int gr = m0 + r;
    for (int c = threadIdx.x; c < K; c += 256)
      sA[r * ldsK + c] = (gr < M) ? A[(size_t)gr * K + c] : 0.0f;
  }
  __syncthreads();
  const int row = lane & 15;
  const int kh  = (lane >> 4) << 1;
  const int mh  = (lane >> 4) << 3;
  v8f acc = {0.f,0.f,0.f,0.f,0.f,0.f,0.f,0.f};
  for (int k = 0; k < K; k += 4) {
    v2f a, b;
    a.x = sA[row * ldsK + k + kh];
    a.y = sA[row * ldsK + k + kh + 1];
    const float* wp = W + (size_t)(k + kh) * Nn + n0 + row;
    b.x = wp[0];
    b.y = wp[Nn];
    acc = __builtin_amdgcn_wmma_f32_16x16x4_f32(false, a, false, b, (short)0, acc, false, false);
  }
  for (int v = 0; v < 8; ++v) {
    int r = m0 + mh + v;
    if (r < M) {
      float x = acc[v];
      if (ACT) x = siluf(x);
      C[(size_t)r * ldc + n0 + row] = x;
    }
  }
}

// ------------------------------------------------------------------
// WMMA fp32 GEMM, small Nn (16/32), K<=128. One wave per 16-row tile.
// grid = ceil(M/128), 256 thr (8 waves).
// ------------------------------------------------------------------
template<bool ACT>
__global__ void k_gemm_ns(const float* __restrict__ A, const float* __restrict__ W,
                          float* __restrict__ C, int M, int K, int Nn, int ldc) {
  __shared__ float sA[8 * 16 * 129];
  const int wave = threadIdx.x >> 5;
  const int lane = threadIdx.x & 31;
  const int m0   = (blockIdx.x * 8 + wave) * 16;
  const int ldsK = K + 1;
  float* myA = sA + (size_t)wave * 16 * ldsK;
  for (int r = 0; r < 16; ++r) {
    int gr = m0 + r;
    for (int c = lane; c < K; c += 32)
      myA[r * ldsK + c] = (gr < M) ? A[(size_t)gr * K + c] : 0.0f;
  }
  __syncthreads();
  const int row = lane & 15;
  const int kh  = (lane >> 4) << 1;
  const int mh  = (lane >> 4) << 3;
  for (int nt = 0; nt < Nn; nt += 16) {
    v8f acc = {0.f,0.f,0.f,0.f,0.f,0.f,0.f,0.f};
    for (int k = 0; k < K; k += 4) {
      v2f a, b;
      a.x = myA[row * ldsK + k + kh];
      a.y = myA[row * ldsK + k + kh + 1];
      const float* wp = W + (size_t)(k + kh) * Nn + nt + row;
      b.x = wp[0];
      b.y = wp[Nn];
      acc = __builtin_amdgcn_wmma_f32_16x16x4_f32(false, a, false, b, (short)0, acc, false, false);
    }
    for (int v = 0; v < 8; ++v) {
      int r = m0 + mh + v;
      if (r < M) {
        float x = acc[v];
        if (ACT) x = siluf(x);
        C[(size_t)r * ldc + nt + row] = x;
      }
    }
  }
}

// ----------------------- elementwise / scatter --------------------
__global__ void k_zero(float* p, int n) {
  int i = blockIdx.x * blockDim.x + threadIdx.x;
  if (i < n) p[i] = 0.0f;
}
__global__ void k_mul(float* out, const float* a, const float* b, int n) {
  int i = blockIdx.x * blockDim.x + threadIdx.x;
  if (i < n) out[i] = a[i] * b[i];
}
__global__ void k_mul_ip(float* a, const float* b, int n) {
  int i = blockIdx.x * blockDim.x + threadIdx.x;
  if (i < n) a[i] *= b[i];
}
__global__ void k_mul_gather(float* x, const float* m, const int* ba, int n) {
  int i = blockIdx.x * blockDim.x + threadIdx.x;
  if (i < n) { int t = i >> 7, c = i & 127; x[i] *= m[(size_t)ba[t] * 128 + c]; }
}
__global__ void k_axpy(float* x, const float* y, int n) {
  int i = blockIdx.x * blockDim.x + threadIdx.x;
  if (i < n) x[i] = (x[i] + y[i]) * INV_SQRT2F;
}
__global__ void k_add3_gather(float* a, const float* b, const float* cg, const int* sw, int n) {
  int i = blockIdx.x * blockDim.x + threadIdx.x;
  if (i < n) { int e = i >> 7, c = i & 127; a[i] = a[i] + b[i] + cg[(size_t)sw[e] * 128 + c]; }
}
__global__ void k_scatter_mul(const float* a, const float* b, const int* tgt, float* out, int n) {
  int i = blockIdx.x * blockDim.x + threadIdx.x;
  if (i < n) { int e = i >> 7, c = i & 127; atomicAdd(&out[(size_t)tgt[e] * 128 + c], a[i] * b[i]); }
}
__global__ void k_concat(const float* h, const int* is, const int* it, const float* third,
                         float* out, int n) {
  int i = blockIdx.x * blockDim.x + threadIdx.x;
  if (i >= n) return;
  int e = i / 384, c = i % 384;
  float v;
  if (c < 128)      v = h[(size_t)is[e] * 128 + c];
  else if (c < 256) v = h[(size_t)it[e] * 128 + (c - 128)];
  else              v = third[(size_t)e * 128 + (c - 256)];
  out[i] = v;
}

// ----------------------- geometry / basis -------------------------
__global__ void k_edge_rbf(const float* v_st, const float* e_st, float* D, float* U,
                           float* V, float* rbf) {
  int e = blockIdx.x, r = threadIdx.x;
  float vx = v_st[e * 3 + 0], vy = v_st[e * 3 + 1], vz = v_st[e * 3 + 2];
  float d = sqrtf(vx * vx + vy * vy + vz * vz);
  if (r == 0) D[e] = d;
  if (r < 3) {
    U[e * 3 + r] = -e_st[e * 3 + r] / d;
    V[e * 3 + r] = -v_st[e * 3 + r] / d;
  }
  const float delta = 6.0f / 127.0f;
  float mu = (float)r * delta;
  float g = __expf(-((d - mu) * (d - mu)) / (2.0f * delta * delta));
  float ds = d / 6.0f;
  float d2 = ds * ds, d4 = d2 * d2, d5 = d4 * ds, d6 = d5 * ds, d7 = d6 * ds;
  float env = (ds < 1.0f) ? (1.0f - 21.0f * d5 + 35.0f * d6 - 15.0f * d7) : 0.0f;
  rbf[(size_t)e * 128 + r] = g * env;
}
__global__ void k_tri_sbf(const float* V, const int* ba, const int* ca, float* sbf, int T) {
  int t = blockIdx.x * blockDim.x + threadIdx.x;
  if (t >= T) return;
  int eb = ba[t], ec = ca[t];
  float ax = V[ec * 3], ay = V[ec * 3 + 1], az = V[ec * 3 + 2];
  float bx = V[eb * 3], by = V[eb * 3 + 1], bz = V[eb * 3 + 2];
  float x = fminf(1.0f, fmaxf(-1.0f, ax * bx + ay * by + az * bz));
  float Pm2 = 1.0f, Pm1 = x;
  sbf[(size_t)t * 7 + 0] = 1.0f;
  sbf[(size_t)t * 7 + 1] = x;
  for (int l = 2; l < 7; ++l) {
    float P = ((2.0f * l - 1.0f) * x * Pm1 - (l - 1.0f) * Pm2) / (float)l;
    sbf[(size_t)t * 7 + l] = P;
    Pm2 = Pm1; Pm1 = P;
  }
}
__global__ void k_h_init(const float* atab, const float* emb, const int* z, const int* batch,
                         float* h, int n, int L) {
  int i = blockIdx.x * blockDim.x + threadIdx.x;
  if (i >= n) return;
  int a = i >> 7, c = i & 127;
  h[i] = atab[(size_t)z[a] * 128 + c] + emb[(size_t)batch[a] * L + c];
}
__global__ void k_cbf3(const float* sbf, const float* radW, const int* ca, float* cbf3, int n) {
  int i = blockIdx.x * blockDim.x + threadIdx.x;
  if (i >= n) return;
  int t = i >> 4, c = i & 15;
  int e = ca[t];
  float acc = 0.0f;
  for (int s = 0; s < 7; ++s)
    acc += sbf[(size_t)t * 7 + s] * radW[(size_t)e * 112 + s * 16 + c];
  cbf3[i] = acc;
}
__global__ void k_cbfo(const float* sbf, const float* W, float* out, int n) {
  int i = blockIdx.x * blockDim.x + threadIdx.x;
  if (i >= n) return;
  int t = i >> 4, c = i & 15;
  float acc = 0.0f;
  for (int s = 0; s < 7; ++s) acc += sbf[(size_t)t * 7 + s] * W[s * 16 + c];
  out[i] = acc;
}

// ----------------------- triplet bilinear -------------------------
__global__ void k_tr_bil(const float* W, float* Wt) { // [16,32,64] -> [32, 16*64]
  int i = blockIdx.x * blockDim.x + threadIdx.x;
  if (i >= 16 * 32 * 64) return;
  int c = i >> 11, e = (i >> 6) & 31, b = i & 63;
  Wt[(size_t)e * 1024 + c * 64 + b] = W[i];
}
__global__ void k_xt(const float* cbf3, const float* Yedge, const int* ba, const int* ca,
                     float* xe, int n) {
  int i = blockIdx.x * blockDim.x + threadIdx.x;
  if (i >= n) return;
  int t = i >> 6, b = i & 63;
  const float* Y = Yedge + (size_t)ba[t] * 1024 + b;
  const float* S = cbf3 + (size_t)t * 16;
  float acc = 0.0f;
#pragma unroll
  for (int c = 0; c < 16; ++c) acc += S[c] * Y[c * 64];
  atomicAdd(&xe[(size_t)ca[t] * 64 + b], acc);
}

// ----------------------- narrow projections -----------------------
template<int J>
__global__ void k_matvec_acc(const float* __restrict__ A, const float* __restrict__ W,
                             float* __restrict__ out, int M) {
  int row = blockIdx.x * 8 + (threadIdx.x >> 5);
  int lane = threadIdx.x & 31;
  if (row >= M) return;
  float acc[J];
#pragma unroll
  for (int j = 0; j < J; ++j) acc[j] = 0.0f;
  for (int k = lane; k < 128; k += 32) {
    float a = A[(size_t)row * 128 + k];
#pragma unroll
    for (int j = 0; j < J; ++j) acc[j] += a * W[k * J + j];
  }
#pragma unroll
  for (int j = 0; j < J; ++j)
    for (int o = 16; o > 0; o >>= 1) acc[j] += __shfl_xor(acc[j], o, 32);
  if (lane == 0)
#pragma unroll
    for (int j = 0; j < J; ++j) out[(size_t)row * J + j] += acc[j];
}

// ----------------------- outputs ----------------------------------
__global__ void k_energy_seg(const float* Et, const int* batch, float* Eb, float* cnt, int N) {
  int n = blockIdx.x * blockDim.x + threadIdx.x;
  if (n >= N) return;
  atomicAdd(&Eb[batch[n]], Et[n]);
  atomicAdd(&cnt[batch[n]], 1.0f);
}
__global__ void k_force_seg(const float* Fst, const float* U, const int* it, float* Ft, int E) {
  int e = blockIdx.x * blockDim.x + threadIdx.x;
  if (e >= E) return;
  float f = Fst[e];
  for (int d = 0; d < 3; ++d) atomicAdd(&Ft[(size_t)it[e] * 3 + d], f * U[e * 3 + d]);
}
__global__ void k_stress(const float* e_st, const float* Sst, const int* ba, const int* ca,
                         const int* idx_s, const int* batch, float* Ssum, float* cntV, int T) {
  int t = blockIdx.x * blockDim.x + threadIdx.x;
  if (t >= T) return;
  int eb = ba[t], ec = ca[t];
  float u[3] = {e_st[eb * 3], e_st[eb * 3 + 1], e_st[eb * 3 + 2]};
  float v[3] = {e_st[ec * 3], e_st[ec * 3 + 1], e_st[ec * 3 + 2]};
  float ru = sqrtf(u[0] * u[0] + u[1] * u[1] + u[2] * u[2]);
  float rv = sqrtf(v[0] * v[0] + v[1] * v[1] + v[2] * v[2]);
  float uh[3], vh[3];
  for (int d = 0; d < 3; ++d) { uh[d] = u[d] / ru; vh[d] = v[d] / rv; }
  float cth = fminf(1.0f, fmaxf(-1.0f, uh[0] * vh[0] + uh[1] * vh[1] + uh[2] * vh[2]));
  float sth = sqrtf(fmaxf(1.0f - cth * cth, 0.0f));
  bool valid = sth > 1e-4f;
  float ss = valid ? sth : 1.0f;
  float dth[9], fn = 0.0f;
  for (int i = 0; i < 3; ++i)
    for (int j = 0; j < 3; ++j) {
      float uu = uh[i] * uh[j], vv = vh[i] * vh[j];
      float sym = 0.5f * (u[i] * v[j] + v[i] * u[j]) / (ru * rv);
      float G = sym - 0.5f * cth * (uu + vv);
      float d = -G / ss;
      dth[i * 3 + j] = d;
      fn += d * d;
    }
  float inv = 1.0f / fmaxf(sqrtf(fn), 1e-6f);
  if (!valid) return;
  float s0 = Sst[(size_t)t * 3], s1 = Sst[(size_t)t * 3 + 1], s2 = Sst[(size_t)t * 3 + 2];
  int b = batch[idx_s[eb]];
  for (int i = 0; i < 3; ++i)
    for (int j = 0; j < 3; ++j) {
      float c = s0 * uh[i] * uh[j] + s1 * vh[i] * vh[j] + s2 * dth[i * 3 + j] * inv;
      atomicAdd(&Ssum[b * 9 + i * 3 + j], c);
    }
  atomicAdd(&cntV[b], 1.0f);
}
__global__ void k_final(const float* h, const float* Eb, const float* cntA, const float* x,
                        const float* Ft, const float* Ssum, const float* cntV,
                        float* out, int N, int B) {
  int i = blockIdx.x * blockDim.x + threadIdx.x;
  int oE = N * 128, oX = oE + B, oF = oX + N * 3, oS = oF + N * 3;
  if (i < N * 128) out[i] = h[i];
  if (i < B) out[oE + i] = Eb[i] / fmaxf(cntA[i], 1.0f);
  if (i < N * 3) {
    float v = x[i] + Ft[i];
    out[oX + i] = v - floorf(v);
    out[oF + i] = Ft[i];
  }
  if (i < B * 9) {
    int ij = i % 9;
    out[oS + i] = Ssum[i] / fmaxf(cntV[i / 9], 1.0f) + ((ij == 0 || ij == 4 || ij == 8) ? 1.0f : 0.0f);
  }
}

// ==================================================================
extern "C" void kernel_launch(void* const* d_in, const int* in_sizes, int n_in,
                              void* d_out, int out_size, void* d_ws, size_t ws_size,
                              hipStream_t stream) {
  auto Fp = [&](int i) { return (const float*)d_in[i]; };
  auto Ip = [&](int i) { return (const int*)d_in[i]; };
  const float *x_ = Fp(0), *v_st = Fp(2), *e_st = Fp(3), *emb = Fp(4), *atab = Fp(5),
      *edgeW = Fp(6), *Wrbf3 = Fp(7), *Wcbf3 = Fp(8), *Wrbfh = Fp(9), *Wrbfo = Fp(10),
      *Wcbfo = Fp(11), *iWxba = Fp(12), *iWtba = Fp(13), *iWtrbf = Fp(14), *iWdown = Fp(15),
      *iWbil = Fp(16), *iWupca = Fp(17), *iWupac = Fp(18), *iResBS = Fp(19), *iResAS = Fp(20),
      *iWhrbf = Fp(21), *iAtomMLP = Fp(22), *iWcat = Fp(23), *iResM = Fp(24), *oWrbfE = Fp(25),
      *oMlpE = Fp(26), *oWE = Fp(27), *oWrbfF = Fp(28), *oMlpF = Fp(29), *oWF = Fp(30),
      *oWcbfS = Fp(31), *oMlpS = Fp(32), *oWS = Fp(33);
  const int *z = Ip(34), *batch = Ip(35), *idx_s = Ip(36), *idx_t = Ip(37), *id_swap = Ip(38),
      *id3_ba = Ip(39), *id3_ca = Ip(40);
  const int N = in_sizes[34], E = in_sizes[36], T = in_sizes[39], B = in_sizes[41];
  const int L = in_sizes[4] / B;
  const int C = 128, R = 128, S = 7, ER = 16, EC = 16, ETT = 32, EBB = 64, NB = 3, NH = 3;

  float* wsf = (float*)d_ws;
  size_t off = 0;
  auto alloc = [&](size_t n) { float* p = wsf + off; off += n; return p; };
  float* Dd = alloc(E);          float* Ud = alloc((size_t)3 * E);  float* Vd = alloc((size_t)3 * E);
  float* rbf = alloc((size_t)E * R);      float* sbf = alloc((size_t)T * S);
  float* h   = alloc((size_t)N * C);      float* hcat = alloc((size_t)E * (2 * C + R));
  float* m   = alloc((size_t)E * C);
  float* rbf3 = alloc((size_t)E * ER);    float* rbfh = alloc((size_t)E * ER);
  float* rbfo = alloc((size_t)E * ER);    float* radW = alloc((size_t)E * S * EC);
  float* cbf3 = alloc((size_t)T * EC);    float* cbfo = alloc((size_t)T * EC);
  float* tA = alloc((size_t)E * C); float* tB = alloc((size_t)E * C); float* tC = alloc((size_t)E * C);
  float* mt = alloc((size_t)E * ETT);     float* Wt = alloc((size_t)EC * ETT * EBB);
  float* Yedge = alloc((size_t)E * EC * EBB);   float* xe = alloc((size_t)E * EBB);
  float* xSa = alloc((size_t)T * C);      float* xSb = alloc((size_t)T * C);
  float* xNacc = alloc((size_t)N * C);    float* xNa = alloc((size_t)N * C); float* xNb = alloc((size_t)N * C);
  float* EtA = alloc(N); float* FstA = alloc(E); float* SstA = alloc((size_t)T * 3);
  float* Eb_ = alloc(B); float* cntA = alloc(B); float* cntV = alloc(B); float* Ssum = alloc((size_t)B * 9);
  float* Ft = alloc((size_t)N * 3);
  (void)ws_size; (void)n_in; (void)out_size; (void)Dd;

  auto cdiv = [](long a, long b) { return (int)((a + b - 1) / b); };
  auto zero = [&](float* p, long n) { k_zero<<<cdiv(n, 256), 256, 0, stream>>>(p, (int)n); };
  auto gemmB = [&](const float* A, const float* W, float* Cp, int M, int K, int Nn, int ldc, bool act) {
    dim3 g(cdiv(M, 16), Nn / 128);
    if (act) k_gemm_nb<true><<<g, 256, 0, stream>>>(A, W, Cp, M, K, Nn, ldc);
    else     k_gemm_nb<false><<<g, 256, 0, stream>>>(A, W, Cp, M, K, Nn, ldc);
  };
  auto gemmS = [&](const float* A, const float* W, float* Cp, int M, int K, int Nn, int ldc, bool act) {
    int g = cdiv(M, 128);
    if (act) k_gemm_ns<true><<<g, 256, 0, stream>>>(A, W, Cp, M, K, Nn, ldc);
    else     k_gemm_ns<false><<<g, 256, 0, stream>>>(A, W, Cp, M, K, Nn, ldc);
  };
  auto axpy = [&](float* x, const float* y, long n) {
    k_axpy<<<cdiv(n, 256), 256, 0, stream>>>(x, y, (int)n);
  };
  auto res_layer = [&](float* xb, const float* W1, const float* W2, int M) {
    gemmB(xb, W1, tB, M, C, C, C, true);
    gemmB(tB, W2, tC, M, C, C, C, true);
    axpy(xb, tC, (long)M * C);
  };
  auto out_block = [&](int i) {
    gemmB(rbfo, oWrbfE + (size_t)i * ER * C, tA, E, ER, C, C, false);
    zero(xNacc, (long)N * C);
    k_scatter_mul<<<cdiv((long)E * C, 256), 256, 0, stream>>>(m, tA, idx_t, xNacc, E * C);
    gemmB(rbfo, oWrbfF + (size_t)i * ER * C, tA, E, ER, C, C, false);
    k_mul<<<cdiv((long)E * C, 256), 256, 0, stream>>>(tB, m, tA, E * C);
    gemmB(cbfo, oWcbfS + (size_t)i * EC * C, xSa, T, EC, C, C, false);
    k_mul_gather<<<cdiv((long)T * C, 256), 256, 0, stream>>>(xSa, m, id3_ba, T * C);
    for (int l = 0; l < NH; ++l) {
      const float* WE = oMlpE + ((size_t)i * NH + l) * C * C;
      const float* WF = oMlpF + ((size_t)i * NH + l) * C * C;
      const float* WS = oMlpS + ((size_t)i * NH + l) * C * C;
      gemmB(l == 0 ? xNacc : (l == 1 ? xNa : xNb), WE, l == 1 ? xNb : xNa, N, C, C, C, true);
      gemmB(l == 1 ? tC : tB, WF, l == 1 ? tB : tC, E, C, C, C, true);
      gemmB(l == 1 ? xSb : xSa, WS, l == 1 ? xSa : xSb, T, C, C, C, true);
    }
    k_matvec_acc<1><<<cdiv(N, 8), 256, 0, stream>>>(xNa, oWE + (size_t)i * C, EtA, N);
    k_matvec_acc<1><<<cdiv(E, 8), 256, 0, stream>>>(tC, oWF + (size_t)i * C, FstA, E);
    k_matvec_acc<3><<<cdiv(T, 8), 256, 0, stream>>>(xSb, oWS + (size_t)i * C * 3, SstA, T);
  };

  // ---------------- geometry, bases, embeddings ----------------
  k_edge_rbf<<<E, 128, 0, stream>>>(v_st, e_st, Dd, Ud, Vd, rbf);
  k_tri_sbf<<<cdiv(T, 256), 256, 0, stream>>>(Vd, id3_ba, id3_ca, sbf, T);
  k_h_init<<<cdiv((long)N * C, 256), 256, 0, stream>>>(atab, emb, z, batch, h, N * C, L);
  gemmS(rbf, Wrbf3, rbf3, E, R, ER, ER, false);
  gemmS(rbf, Wrbfh, rbfh, E, R, ER, ER, false);
  gemmS(rbf, Wrbfo, rbfo, E, R, ER, ER, false);
  for (int s = 0; s < S; ++s)
    gemmS(rbf, Wcbf3 + (size_t)s * R * EC, radW + (size_t)s * EC, E, R, EC, S * EC, false);
  k_cbf3<<<cdiv((long)T * EC, 256), 256, 0, stream>>>(sbf, radW, id3_ca, cbf3, T * EC);
  k_cbfo<<<cdiv((long)T * EC, 256), 256, 0, stream>>>(sbf, Wcbfo, cbfo, T * EC);
  k_concat<<<cdiv((long)E * 384, 256), 256, 0, stream>>>(h, idx_s, idx_t, rbf, hcat, E * 384);
  gemmB(hcat, edgeW, m, E, 3 * C, C, C, true);
  zero(EtA, N); zero(FstA, E); zero(SstA, (long)T * 3);
  out_block(0);

  // ---------------- interaction blocks ----------------
  for (int i = 0; i < NB; ++i) {
    gemmB(m, iWxba + (size_t)i * C * C, tA, E, C, C, C, true);               // xba
    gemmB(m, iWtba + (size_t)i * C * C, tB, E, C, C, C, true);
    gemmB(rbf3, iWtrbf + (size_t)i * ER * C, tC, E, ER, C, C, false);
    k_mul_ip<<<cdiv((long)E * C, 256), 256, 0, stream>>>(tB, tC, E * C);     // t1
    gemmS(tB, iWdown + (size_t)i * C * ETT, mt, E, C, ETT, ETT, true);
    k_tr_bil<<<cdiv(EC * ETT * EBB, 256), 256, 0, stream>>>(iWbil + (size_t)i * EC * ETT * EBB, Wt);
    gemmB(mt, Wt, Yedge, E, ETT, EC * EBB, EC * EBB, false);
    zero(xe, (long)E * EBB);
    k_xt<<<cdiv((long)T * EBB, 256), 256, 0, stream>>>(cbf3, Yedge, id3_ba, id3_ca, xe, T * EBB);
    gemmB(xe, iWupca + (size_t)i * EBB * C, tB, E, EBB, C, C, true);
    gemmB(xe, iWupac + (size_t)i * EBB * C, tC, E, EBB, C, C, true);
    k_add3_gather<<<cdiv((long)E * C, 256), 256, 0, stream>>>(tA, tB, tC, id_swap, E * C);
    res_layer(tA, iResBS + ((size_t)i * 2 + 0) * C * C, iResBS + ((size_t)i * 2 + 1) * C * C, E);
    axpy(m, tA, (long)E * C);
    for (int l = 0; l < 2; ++l)
      res_layer(m, iResAS + (((size_t)i * 2 + l) * 2 + 0) * C * C,
                   iResAS + (((size_t)i * 2 + l) * 2 + 1) * C * C, E);
    // atom update
    gemmB(rbfh, iWhrbf + (size_t)i * ER * C, tA, E, ER, C, C, false);
    zero(xNacc, (long)N * C);
    k_scatter_mul<<<cdiv((long)E * C, 256), 256, 0, stream>>>(m, tA, idx_t, xNacc, E * C);
    for (int l = 0; l < NH; ++l)
      gemmB(l == 0 ? xNacc : (l == 1 ? xNa : xNb), iAtomMLP + ((size_t)i * NH + l) * C * C,
            l == 1 ? xNb : xNa, N, C, C, C, true);
    axpy(h, xNa, (long)N * C);
    // concat / edge re-embedding
    k_concat<<<cdiv((long)E * 384, 256), 256, 0, stream>>>(h, idx_s, idx_t, m, hcat, E * 384);
    gemmB(hcat, iWcat + (size_t)i * (3 * C) * C, tA, E, 3 * C, C, C, true);
    res_layer(tA, iResM + ((size_t)i * 2 + 0) * C * C, iResM + ((size_t)i * 2 + 1) * C * C, E);
    axpy(m, tA, (long)E * C);
    out_block(i + 1);
  }

  // ---------------- outputs ----------------
  zero(Eb_, B); zero(cntA, B); zero(cntV, B); zero(Ssum, (long)B * 9); zero(Ft, (long)N * 3);
  k_energy_seg<<<cdiv(N, 256), 256, 0, stream>>>(EtA, batch, Eb_, cntA, N);
  k_force_seg<<<cdiv(E, 256), 256, 0, stream>>>(FstA, Ud, idx_t, Ft, E);
  k_stress<<<cdiv(T, 256), 256, 0, stream>>>(e_st, SstA, id3_ba, id3_ca, idx_s, batch, Ssum, cntV, T);
  k_final<<<cdiv((long)N * C, 256), 256, 0, stream>>>(h, Eb_, cntA, x_, Ft, Ssum, cntV,
                                                      (float*)d_out, N, B);
}